// UnifiedSelfAttentionQLSTM_65481071398196
// MI455X (gfx1250) — compile-verified
//
#include <hip/hip_runtime.h>
#include <math.h>

typedef float v2f __attribute__((ext_vector_type(2)));
typedef float v8f __attribute__((ext_vector_type(8)));

#define NQ   4
#define DMODEL 256
#define NHEAD  4
#define HDIM   64
#define NTAG   64
#define NB     16
#define NT     512
#define DQ   (DMODEL + NQ)   /* 260 */

static __device__ __forceinline__ v8f wmma4(v2f a, v2f b, v8f c) {
    // V_WMMA_F32_16X16X4_F32 : D = A(16x4) * B(4x16) + C(16x16), fp32 throughout
    return __builtin_amdgcn_wmma_f32_16x16x4_f32(
        /*neg_a=*/false, a, /*neg_b=*/false, b,
        /*c_mod=*/(short)0, c, /*reuse_a=*/false, /*reuse_b=*/false);
}

// ---------------------------------------------------------------- embedding
__global__ void k_embed(const int* __restrict__ sent,
                        const float* __restrict__ emb,
                        float* __restrict__ x) {
    int row = blockIdx.x;                 // 0 .. B*T-1
    int tok = sent[row];
    const float4* src = (const float4*)(emb + (size_t)tok * DMODEL);
    float4*       dst = (float4*)(x + (size_t)row * DMODEL);
    dst[threadIdx.x] = src[threadIdx.x];  // 64 threads x 16B = 1KB row
}

// ------------------------------------------------- C = A @ W^T + bias (WMMA)
// A: [M,K] row-major, W: [N,K] row-major (so B-matrix B[k][n] = W[n][k])
__global__ void k_gemm_bias(const float* __restrict__ A,
                            const float* __restrict__ W,
                            const float* __restrict__ bias,
                            float* __restrict__ C,
                            int M, int N, int K) {
    int lane = threadIdx.x & 31;
    int wave = threadIdx.x >> 5;
    int tile = blockIdx.x * 4 + wave;
    int ntiles = N >> 4;
    int m0 = (tile / ntiles) << 4;
    int n0 = (tile % ntiles) << 4;
    int mr = lane & 15;
    int koff = (lane < 16) ? 0 : 2;       // A frag: lanes 0-15 -> K{0,1}, 16-31 -> K{2,3}
    const float* arow = A + (size_t)(m0 + mr) * K + koff;
    const float* brow = W + (size_t)(n0 + mr) * K + koff;  // B frag mirrors A (W^T)
    v8f c = {0.f,0.f,0.f,0.f,0.f,0.f,0.f,0.f};
    for (int k0 = 0; k0 < K; k0 += 4) {
        v2f a = *(const v2f*)(arow + k0);
        v2f b = *(const v2f*)(brow + k0);
        __builtin_prefetch(arow + k0 + 64, 0, 1);   // global_prefetch_b8
        c = wmma4(a, b, c);
    }
    int rbase = m0 + ((lane < 16) ? 0 : 8);         // C frag: VGPR r -> M=r / M=r+8
    int col = n0 + mr;
    float bv = bias[col];
#pragma unroll
    for (int r = 0; r < 8; ++r)
        C[(size_t)(rbase + r) * N + col] = c[r] + bv;
}

// ------------------------------------------------------------- attention
// one wave per (b, h, 16-query tile); scores row (16x512) staged in LDS
#define SDIM 516   // pad so s[m][*] rows land on distinct banks (4*m mod 64)
__global__ void k_attn(const float* __restrict__ q,
                       const float* __restrict__ k,
                       const float* __restrict__ v,
                       float* __restrict__ ctx) {
    __shared__ float s[16 * SDIM];
    int lane = threadIdx.x;
    int qt = blockIdx.x & 31;           // T/16 = 32
    int h  = (blockIdx.x >> 5) & 3;
    int b  = blockIdx.x >> 7;
    int q0 = qt << 4;
    int mr = lane & 15;
    int hi = (lane < 16) ? 0 : 1;
    int koff = hi * 2;
    const float* qbase = q + (size_t)(b * NT + q0 + mr) * DMODEL + h * HDIM + koff;
    const float* kbase = k + (size_t)(b * NT) * DMODEL + h * HDIM;
    const float* vbase = v + (size_t)(b * NT) * DMODEL + h * HDIM;

    // S = (Q K^T) / sqrt(64)
    for (int kt = 0; kt < 32; ++kt) {
        v8f c = {0.f,0.f,0.f,0.f,0.f,0.f,0.f,0.f};
        const float* krow = kbase + (size_t)(kt * 16 + mr) * DMODEL + koff;
#pragma unroll
        for (int kk = 0; kk < HDIM; kk += 4) {
            v2f a  = *(const v2f*)(qbase + kk);
            v2f bb = *(const v2f*)(krow + kk);
            c = wmma4(a, bb, c);
        }
        int srow = hi * 8;
#pragma unroll
        for (int r = 0; r < 8; ++r)
            s[(srow + r) * SDIM + kt * 16 + mr] = c[r] * 0.125f;
    }
    asm volatile("s_wait_dscnt 0" ::: "memory");

    // row softmax (single wave, shfl reductions)
    for (int r = 0; r < 16; ++r) {
        float* row = s + r * SDIM;
        float mx = -3.4e38f;
        for (int i = lane; i < NT; i += 32) mx = fmaxf(mx, row[i]);
#pragma unroll
        for (int off = 16; off > 0; off >>= 1) mx = fmaxf(mx, __shfl_xor(mx, off, 32));
        float sum = 0.f;
        for (int i = lane; i < NT; i += 32) {
            float e = __expf(row[i] - mx); row[i] = e; sum += e;
        }
#pragma unroll
        for (int off = 16; off > 0; off >>= 1) sum += __shfl_xor(sum, off, 32);
        float inv = 1.f / sum;
        for (int i = lane; i < NT; i += 32) row[i] *= inv;
    }
    asm volatile("s_wait_dscnt 0" ::: "memory");

    // O = P @ V   (16 x 64, K = 512)
    const float* prow = s + mr * SDIM + koff;
    for (int nt = 0; nt < 4; ++nt) {
        v8f c = {0.f,0.f,0.f,0.f,0.f,0.f,0.f,0.f};
        int n = nt * 16 + mr;
        for (int k0 = 0; k0 < NT; k0 += 4) {
            v2f a = *(const v2f*)(prow + k0);
            v2f bb;
            bb.x = vbase[(size_t)(k0 + koff) * DMODEL + n];
            bb.y = vbase[(size_t)(k0 + koff + 1) * DMODEL + n];
            c = wmma4(a, bb, c);
        }
        int rbase = q0 + hi * 8;
#pragma unroll
        for (int r = 0; r < 8; ++r)
            ctx[(size_t)(b * NT + rbase + r) * DMODEL + h * HDIM + n] = c[r];
    }
}

// ------------------------------------------------------------- QLSTM
// persistent single block: weights + recurrent state in LDS; wave0 does the
// 16(batch) x 16(gate-out) x 260 GEMM per step via WMMA; 64 threads simulate
// the 4-qubit circuit (RX layer composed analytically: RX(a)RX(th)=RX(a+th)).
#define WS 268    // pad 260 -> 268 so row stride hits distinct banks (12*m mod 64)
__global__ void k_qlstm(const float* __restrict__ xin,
                        const float* __restrict__ Wf, const float* __restrict__ bf, const float* __restrict__ thf,
                        const float* __restrict__ Wi, const float* __restrict__ bi, const float* __restrict__ thi,
                        const float* __restrict__ Wg, const float* __restrict__ bg, const float* __restrict__ thg,
                        const float* __restrict__ Wo2, const float* __restrict__ bo2, const float* __restrict__ tho,
                        float* __restrict__ hout) {
    __shared__ float Wl[16 * WS];
    __shared__ float comb[16 * WS];
    __shared__ float gates[16 * 16];
    __shared__ float hx[64], cx[64];
    __shared__ float bias[16], th[16];
    int tid = threadIdx.x;

    for (int i = tid; i < 16 * DQ; i += 256) {
        int r = i / DQ, kk = i % DQ;
        const float* Wsel = (r < 4) ? Wf : (r < 8) ? Wi : (r < 12) ? Wg : Wo2;
        Wl[r * WS + kk] = Wsel[(r & 3) * DQ + kk];
    }
    if (tid < 16) {
        const float* bsel = (tid < 4) ? bf : (tid < 8) ? bi : (tid < 12) ? bg : bo2;
        const float* tsel = (tid < 4) ? thf : (tid < 8) ? thi : (tid < 12) ? thg : tho;
        bias[tid] = bsel[tid & 3];
        th[tid]   = tsel[tid & 3];
    }
    if (tid < 64) { hx[tid] = 0.f; cx[tid] = 0.f; }
    __syncthreads();

    int lane = tid & 31;
    int mr = lane & 15;
    int koff = (lane < 16) ? 0 : 2;

    for (int t = 0; t < NT; ++t) {
        // comb = [x_t | hx]
        for (int i = tid; i < 16 * DMODEL; i += 256) {
            int bb = i >> 8, kk = i & 255;
            comb[bb * WS + kk] = xin[(size_t)(bb * NT + t) * DMODEL + kk];
        }
        if (tid < 64) comb[(tid >> 2) * WS + DMODEL + (tid & 3)] = hx[tid];
        __syncthreads();

        if (tid < 32) {   // wave 0: gates_raw = comb @ Wl^T + bias  (16x16, K=260)
            v8f c = {0.f,0.f,0.f,0.f,0.f,0.f,0.f,0.f};
            const float* arow = comb + mr * WS + koff;
            const float* brow = Wl   + mr * WS + koff;
            for (int k0 = 0; k0 < DQ; k0 += 4) {
                v2f a  = *(const v2f*)(arow + k0);
                v2f bv = *(const v2f*)(brow + k0);
                c = wmma4(a, bv, c);
            }
            int m = (lane < 16) ? 0 : 8;
#pragma unroll
            for (int r = 0; r < 8; ++r)
                gates[(m + r) * 16 + mr] = c[r] + bias[mr];
        }
        __syncthreads();

        if (tid < 64) {   // (batch, gate) -> 4-qubit circuit
            int bb = tid >> 2, g = tid & 3;
            float sr[16], si[16];
#pragma unroll
            for (int i = 0; i < 16; ++i) { sr[i] = 0.f; si[i] = 0.f; }
            sr[0] = 1.f;
#pragma unroll
            for (int w = 0; w < 4; ++w) {   // RX(angle_w + theta_w) on wire w
                float half = 0.5f * (gates[bb * 16 + g * 4 + w] + th[g * 4 + w]);
                float cw = cosf(half), sw = sinf(half);
                int bit = 8 >> w;
#pragma unroll
                for (int i = 0; i < 16; ++i) {
                    if ((i & bit) == 0) {
                        int j = i | bit;
                        float ar = sr[i], ai = si[i], br = sr[j], bi2 = si[j];
                        sr[i] = cw * ar + sw * bi2;  si[i] = cw * ai - sw * br;
                        sr[j] = cw * br + sw * ai;   si[j] = cw * bi2 - sw * ar;
                    }
                }
            }
#pragma unroll
            for (int p = 0; p < 4; ++p) {   // CNOT ring (0,1)(1,2)(2,3)(3,0)
                int cb = 8 >> p;
                int tb = 8 >> ((p + 1) & 3);
#pragma unroll
                for (int i = 0; i < 16; ++i) {
                    if ((i & cb) && !(i & tb)) {
                        int j = i | tb;
                        float tr = sr[i]; sr[i] = sr[j]; sr[j] = tr;
                        float ti = si[i]; si[i] = si[j]; si[j] = ti;
                    }
                }
            }
            float outv[4] = {0.f, 0.f, 0.f, 0.f};
#pragma unroll
            for (int i = 0; i < 16; ++i) {
                float pz = sr[i] * sr[i] + si[i] * si[i];
#pragma unroll
                for (int w = 0; w < 4; ++w)
                    outv[w] += (i & (8 >> w)) ? -pz : pz;
            }
#pragma unroll
            for (int w = 0; w < 4; ++w) {
                float a = (g == 2) ? tanhf(outv[w])
                                   : 1.f / (1.f + __expf(-outv[w]));
                gates[bb * 16 + g * 4 + w] = a;
            }
        }
        __syncthreads();

        if (tid < 64) {   // cell update
            int bb = tid >> 2, w = tid & 3;
            float f  = gates[bb * 16 + 0  + w];
            float ii = gates[bb * 16 + 4  + w];
            float gg = gates[bb * 16 + 8  + w];
            float oo = gates[bb * 16 + 12 + w];
            float cn = f * cx[tid] + ii * gg;
            float hn = oo * tanhf(cn);
            cx[tid] = cn; hx[tid] = hn;
            hout[(size_t)(bb * NT + t) * NQ + w] = hn;
        }
        __syncthreads();
    }
}

// ------------------------------------------------- tag head + log_softmax
__global__ void k_tag(const float* __restrict__ hseq,
                      const float* __restrict__ Wt,
                      const float* __restrict__ bt,
                      float* __restrict__ out) {
    __shared__ float red[NTAG];
    int row = blockIdx.x;
    int tg = threadIdx.x;
    const float* hr = hseq + (size_t)row * NQ;
    float logit = bt[tg];
#pragma unroll
    for (int j = 0; j < NQ; ++j) logit += hr[j] * Wt[tg * NQ + j];
    red[tg] = logit; __syncthreads();
    for (int off = 32; off > 0; off >>= 1) {
        if (tg < off) red[tg] = fmaxf(red[tg], red[tg + off]);
        __syncthreads();
    }
    float mx = red[0]; __syncthreads();
    red[tg] = __expf(logit - mx); __syncthreads();
    for (int off = 32; off > 0; off >>= 1) {
        if (tg < off) red[tg] += red[tg + off];
        __syncthreads();
    }
    out[(size_t)row * NTAG + tg] = logit - mx - logf(red[0]);
}

// ----------------------------------------------------------------- launch
extern "C" void kernel_launch(void* const* d_in, const int* in_sizes, int n_in,
                              void* d_out, int out_size, void* d_ws, size_t ws_size,
                              hipStream_t stream) {
    const int*   sent = (const int*)  d_in[0];
    const float* emb  = (const float*)d_in[1];
    const float* Wq = (const float*)d_in[2];  const float* bq = (const float*)d_in[3];
    const float* Wk = (const float*)d_in[4];  const float* bk = (const float*)d_in[5];
    const float* Wv = (const float*)d_in[6];  const float* bv = (const float*)d_in[7];
    const float* Wo = (const float*)d_in[8];  const float* bo = (const float*)d_in[9];
    const float* Wf = (const float*)d_in[10]; const float* bf = (const float*)d_in[11]; const float* thf = (const float*)d_in[12];
    const float* Wi = (const float*)d_in[13]; const float* bi = (const float*)d_in[14]; const float* thi = (const float*)d_in[15];
    const float* Wg = (const float*)d_in[16]; const float* bg = (const float*)d_in[17]; const float* thg = (const float*)d_in[18];
    const float* Wo2= (const float*)d_in[19]; const float* bo2= (const float*)d_in[20]; const float* tho= (const float*)d_in[21];
    const float* Wt = (const float*)d_in[22]; const float* bt = (const float*)d_in[23];
    float* out = (float*)d_out;
    float* ws  = (float*)d_ws;

    const size_t MTOK = (size_t)NB * NT;    // 8192 rows
    const size_t SZ   = MTOK * DMODEL;      // 2M floats per activation buffer
    float* x    = ws;
    float* qb   = x    + SZ;
    float* kb   = qb   + SZ;
    float* vb   = kb   + SZ;
    float* ctx  = vb   + SZ;
    float* aout = ctx  + SZ;
    float* hseq = aout + SZ;                // B*T*NQ

    k_embed<<<(int)MTOK, 64, 0, stream>>>(sent, emb, x);

    int blocks = (int)((MTOK / 16) * (DMODEL / 16)) / 4;  // 4 waves/block
    k_gemm_bias<<<blocks, 128, 0, stream>>>(x, Wq, bq, qb, (int)MTOK, DMODEL, DMODEL);
    k_gemm_bias<<<blocks, 128, 0, stream>>>(x, Wk, bk, kb, (int)MTOK, DMODEL, DMODEL);
    k_gemm_bias<<<blocks, 128, 0, stream>>>(x, Wv, bv, vb, (int)MTOK, DMODEL, DMODEL);

    k_attn<<<NB * NHEAD * (NT / 16), 32, 0, stream>>>(qb, kb, vb, ctx);

    k_gemm_bias<<<blocks, 128, 0, stream>>>(ctx, Wo, bo, aout, (int)MTOK, DMODEL, DMODEL);

    k_qlstm<<<1, 256, 0, stream>>>(aout, Wf, bf, thf, Wi, bi, thi,
                                   Wg, bg, thg, Wo2, bo2, tho, hseq);

    k_tag<<<(int)MTOK, NTAG, 0, stream>>>(hseq, Wt, bt, out);
}